// SCL_75419625717886
// MI455X (gfx1250) — compile-verified
//
#include <hip/hip_runtime.h>

typedef float v2f __attribute__((ext_vector_type(2)));
typedef float v8f __attribute__((ext_vector_type(8)));

#define NB 8
#define NH 16
#define NS 4096
#define ND 128
#define WAVES_PER_BLOCK 8
#define EPS 1e-8f

// One wave computes the 16x16 f32 Gram matrix of the 16x128 head tile at one
// (b, s) position via V_WMMA_F32_16X16X4_F32 accumulation over K=128, then
// extracts dots-vs-anchor (column 0) and squared norms (diagonal) to form the
// cosine-similarity partial sum.
__global__ __launch_bounds__(256) void SCL_gram_cossim_kernel(
    const float* __restrict__ heads, float* __restrict__ acc) {
  const int lane = threadIdx.x & 31;
  const int wave = threadIdx.x >> 5;
  const int idx = blockIdx.x * WAVES_PER_BLOCK + wave;  // flat (b, s)
  const int b = idx >> 12;          // idx / 4096
  const int s = idx & (NS - 1);     // idx % 4096

  // WMMA f32 A(16x4) / B(4x16) layout (identical for symmetric Gram input):
  //   lanes 0-15 : row M = lane,    VGPR0 = K0, VGPR1 = K1
  //   lanes 16-31: row M = lane-16, VGPR0 = K2, VGPR1 = K3
  const int head = lane & 15;
  const int koff = (lane >> 4) * 2;

  const float* row =
      heads + (((size_t)b * NH + head) * NS + s) * (size_t)ND + koff;

  // Two accumulators to break the WMMA->WMMA C-chain dependency in half.
  v8f c0 = {};
  v8f c1 = {};
#pragma unroll
  for (int k = 0; k < ND; k += 8) {
    v2f a0 = *(const v2f*)(row + k);
    v2f a1 = *(const v2f*)(row + k + 4);
    c0 = __builtin_amdgcn_wmma_f32_16x16x4_f32(
        false, a0, false, a0, (short)0, c0, false, false);
    c1 = __builtin_amdgcn_wmma_f32_16x16x4_f32(
        false, a1, false, a1, (short)0, c1, false, false);
  }
  v8f c = c0 + c1;  // G tile: lanes 0-15 rows M=0..7 (VGPR r -> M=r), N=lane;
                    //         lanes 16-31 rows M=8..15, N=lane-16.

  // G[0][0] = ||anchor||^2 lives at lane 0, VGPR 0.
  const float anchor_n2 = __shfl(c[0], 0, 32);

  // For VGPR r: head r's dot is G[r][0] (lane 0), head r+8's dot is G[r+8][0]
  // (lane 16). Diagonals: G[r][r] at lane r, G[r+8][r+8] at lane r+24.
  float simsum = 0.0f;
#pragma unroll
  for (int r = 0; r < 8; ++r) {
    const float dot_lo = __shfl(c[r], 0, 32);
    const float dot_hi = __shfl(c[r], 16, 32);
    const float n2_lo = __shfl(c[r], r, 32);
    const float n2_hi = __shfl(c[r], r + 24, 32);
    if (r >= 1) {  // heads 1..7 are "others"
      const float denom = fmaxf(sqrtf(n2_lo * anchor_n2), EPS);
      simsum += dot_lo / denom;
    }
    if (r <= 6) {  // heads 8..14 are "others"; head 15 excluded
      const float denom = fmaxf(sqrtf(n2_hi * anchor_n2), EPS);
      simsum += dot_hi / denom;
    }
  }

  // Per-(b,s) contribution to sum of (1 - sim): 14 heads.
  __shared__ float blk;
  if (threadIdx.x == 0) blk = 0.0f;
  __syncthreads();
  if (lane == 0) atomicAdd(&blk, 14.0f - simsum);
  __syncthreads();
  if (threadIdx.x == 0) atomicAdd(acc, blk);
}

__global__ void SCL_finalize_kernel(const float* __restrict__ acc,
                                    float* __restrict__ out) {
  out[0] = acc[0] * (1.0f / ((float)NB * 14.0f * (float)NS));
}

extern "C" void kernel_launch(void* const* d_in, const int* in_sizes, int n_in,
                              void* d_out, int out_size, void* d_ws,
                              size_t ws_size, hipStream_t stream) {
  const float* heads = (const float*)d_in[0];
  float* out = (float*)d_out;
  float* acc = (float*)d_ws;

  hipMemsetAsync(acc, 0, sizeof(float), stream);

  const int total_waves = NB * NS;                     // 32768
  const int blocks = total_waves / WAVES_PER_BLOCK;    // 4096
  SCL_gram_cossim_kernel<<<blocks, WAVES_PER_BLOCK * 32, 0, stream>>>(heads,
                                                                      acc);
  SCL_finalize_kernel<<<1, 1, 0, stream>>>(acc, out);
}